// GCBlock_15032385536630
// MI455X (gfx1250) — compile-verified
//
#include <hip/hip_runtime.h>
#include <hip/hip_bf16.h>

// Fused GCBlock for MI455X (gfx1250, wave32).
// - fp32 WMMA (V_WMMA_F32_16X16X4_F32) for both GEMMs (full precision, K small).
// - Per-pair fusion: never materialize inter[800000x256] / i1[800000x192] in HBM.
// - Segment sums via global fp32 atomics (accumulators are L2-resident, 51 MB).

typedef float v2f __attribute__((ext_vector_type(2)));
typedef float v8f __attribute__((ext_vector_type(8)));

#define FDIM   64
#define NATOMS 50000
#define NPAIRS 800000

static __device__ __forceinline__ v8f wmma_f32(v2f a, v2f b, v8f c) {
  // (neg_a, A, neg_b, B, c_mod, C, reuse_a, reuse_b)
  return __builtin_amdgcn_wmma_f32_16x16x4_f32(false, a, false, b, (short)0, c,
                                               false, false);
}

// A fragment (16x4 tile of row-major MxK matrix in LDS, row stride lda floats).
// ISA layout: lanes 0-15 hold M=lane, K=k0..k0+1 in v[0..1]; lanes 16-31 hold K=k0+2..k0+3.
static __device__ __forceinline__ v2f a_frag(const float* A, int lda, int lane, int k0) {
  int row = lane & 15;
  int kh  = (lane >> 4) << 1;             // 0 or 2
  const float* p = A + row * lda + k0 + kh;
  v2f a; a.x = p[0]; a.y = p[1]; return a;
}

// B fragment (4x16 tile of row-major KxN matrix in global, row stride ldb floats).
// Layout mirrors A: VGPR0 = B[k0+kh][n0+col], VGPR1 = B[k0+kh+1][n0+col].
static __device__ __forceinline__ v2f b_frag(const float* __restrict__ B, int ldb,
                                             int lane, int k0, int n0) {
  int col = lane & 15;
  int kh  = (lane >> 4) << 1;
  const float* p = B + (size_t)(k0 + kh) * ldb + n0 + col;
  v2f b; b.x = p[0]; b.y = p[ldb]; return b;
}

// ---------------------------------------------------------------------------
// Kernel 1: h = relu(relu(p1 @ W0 + b0) @ W1 + b1)   [16 atoms per 128-thr block]
// ---------------------------------------------------------------------------
__global__ __launch_bounds__(128) void atom_mlp_kernel(
    const float* __restrict__ p1, const float* __restrict__ w0,
    const float* __restrict__ b0, const float* __restrict__ w1,
    const float* __restrict__ b1, float* __restrict__ h) {
  __shared__ float lA[16 * FDIM];
  __shared__ float lH[16 * FDIM];
  const int tid = threadIdx.x, lane = tid & 31, wave = tid >> 5;
  const int arow0 = blockIdx.x * 16;

  for (int k = 0; k < 8; ++k) {
    int e = tid + k * 128;                         // 1024 contiguous floats
    lA[e] = p1[(size_t)arow0 * FDIM + e];
  }
  __syncthreads();

  { // layer 0: wave w owns output columns [16w, 16w+16)
    const int n0 = wave * 16;
    v8f acc = {};
    for (int kk = 0; kk < 16; ++kk) {
      v2f a = a_frag(lA, FDIM, lane, kk * 4);
      v2f b = b_frag(w0, FDIM, lane, kk * 4, n0);
      acc = wmma_f32(a, b, acc);
    }
    const int col = lane & 15, rbase = (lane >> 4) * 8;
    const float bias = b0[n0 + col];
    for (int r = 0; r < 8; ++r) {
      float v = acc[r] + bias;
      lH[(rbase + r) * FDIM + n0 + col] = v > 0.f ? v : 0.f;
    }
  }
  __syncthreads();

  { // layer 1
    const int n0 = wave * 16;
    v8f acc = {};
    for (int kk = 0; kk < 16; ++kk) {
      v2f a = a_frag(lH, FDIM, lane, kk * 4);
      v2f b = b_frag(w1, FDIM, lane, kk * 4, n0);
      acc = wmma_f32(a, b, acc);
    }
    const int col = lane & 15, rbase = (lane >> 4) * 8;
    const float bias = b1[n0 + col];
    for (int r = 0; r < 8; ++r) {
      float v = acc[r] + bias;
      h[(size_t)(arow0 + rbase + r) * FDIM + n0 + col] = v > 0.f ? v : 0.f;
    }
  }
}

// ---------------------------------------------------------------------------
// Kernel 2: fused pair pipeline, 16 pairs per 128-thread (4-wave) block.
//   gather h[i]|h[j] -> inter = A@pi_w+pi_b -> i1 = inter·basis
//   -> i1o = relu(i1@ii1_w) -> atomic scatter into p1acc / p3acc
// ---------------------------------------------------------------------------
__global__ __launch_bounds__(128) void pair_kernel(
    const float* __restrict__ h, const float* __restrict__ p3,
    const float* __restrict__ basis, const float* __restrict__ diff,
    const float* __restrict__ piw, const float* __restrict__ pib,
    const float* __restrict__ iiw, const long long* __restrict__ ind2,
    float* __restrict__ p1acc, float* __restrict__ p3acc) {
  __shared__ float lA[16 * 128];       //  8 KB  gathered h rows (i | j)
  __shared__ float lInter[16 * 256];   // 16 KB
  __shared__ float lI1[16 * 64];       //  4 KB
  __shared__ float lOut[16 * 192];     // 12 KB  [i1_1 | i1_2 | i1_3]
  __shared__ int   lIdxI[16], lIdxJ[16];
  __shared__ float lBasis[16 * 4];
  __shared__ float lDiff[16 * 3];

  const int tid = threadIdx.x, lane = tid & 31, wave = tid >> 5;
  const int p0 = blockIdx.x * 16;

  if (tid < 16) {
    lIdxI[tid] = (int)ind2[(size_t)(p0 + tid) * 2 + 0];
    lIdxJ[tid] = (int)ind2[(size_t)(p0 + tid) * 2 + 1];
  }
  if (tid < 64) lBasis[tid] = basis[(size_t)p0 * 4 + tid];
  if (tid < 48) lDiff[tid]  = diff[(size_t)p0 * 3 + tid];
  __syncthreads();

  // gather 16 x 128 concat rows from h (L2-resident, 12.8 MB)
  for (int k = 0; k < 16; ++k) {
    int e = tid + k * 128;
    int pr = e >> 7, c = e & 127;
    int atom = (c < FDIM) ? lIdxI[pr] : lIdxJ[pr];
    lA[e] = h[(size_t)atom * FDIM + (c & 63)];
  }
  __syncthreads();

  // GEMM1: inter[16x256] = lA[16x128] @ piw[128x256] + pib  (4 tiles/wave)
  for (int tt = 0; tt < 4; ++tt) {
    const int n0 = (wave + tt * 4) * 16;
    v8f acc = {};
    for (int kk = 0; kk < 32; ++kk) {
      v2f a = a_frag(lA, 128, lane, kk * 4);
      v2f b = b_frag(piw, 256, lane, kk * 4, n0);
      acc = wmma_f32(a, b, acc);
    }
    const int col = lane & 15, rbase = (lane >> 4) * 8;
    const float bias = pib[n0 + col];
    for (int r = 0; r < 8; ++r)
      lInter[(rbase + r) * 256 + n0 + col] = acc[r] + bias;
  }
  __syncthreads();

  // basis contraction: i1[p][c] = sum_b inter[p][c*4+b] * basis[p][b]
  for (int k = 0; k < 8; ++k) {
    int e = tid + k * 128;               // 1024 = 16 pairs * 64 channels
    int pr = e >> 6, c = e & 63;
    const float* ip = &lInter[pr * 256 + c * 4];
    const float* bp = &lBasis[pr * 4];
    lI1[e] = ip[0] * bp[0] + ip[1] * bp[1] + ip[2] * bp[2] + ip[3] * bp[3];
  }
  __syncthreads();

  // GEMM2: i1o[16x192] = relu(lI1[16x64] @ iiw[64x192])  (3 tiles/wave)
  for (int tt = 0; tt < 3; ++tt) {
    const int n0 = (wave + tt * 4) * 16;
    v8f acc = {};
    for (int kk = 0; kk < 16; ++kk) {
      v2f a = a_frag(lI1, 64, lane, kk * 4);
      v2f b = b_frag(iiw, 192, lane, kk * 4, n0);
      acc = wmma_f32(a, b, acc);
    }
    const int col = lane & 15, rbase = (lane >> 4) * 8;
    for (int r = 0; r < 8; ++r) {
      float v = acc[r];
      lOut[(rbase + r) * 192 + n0 + col] = v > 0.f ? v : 0.f;
    }
  }
  __syncthreads();

  // scatter: p1_new[idx_i] += i1_2
  for (int k = 0; k < 8; ++k) {
    int e = tid + k * 128;
    int pr = e >> 6, c = e & 63;
    atomicAdd(&p1acc[(size_t)lIdxI[pr] * FDIM + c], lOut[pr * 192 + 64 + c]);
  }
  // scatter: p3_new[idx_i] += p3[idx_j]*i1_3 + diff*i1_1
  for (int k = 0; k < 24; ++k) {
    int e = tid + k * 128;               // 3072 = 16 pairs * 3 * 64
    int pr = e / 192, r = e - pr * 192;
    int x = r >> 6, c = r & 63;
    float v = p3[(size_t)lIdxJ[pr] * 192 + r] * lOut[pr * 192 + 128 + c]
            + lDiff[pr * 3 + x]               * lOut[pr * 192 + c];
    atomicAdd(&p3acc[(size_t)lIdxI[pr] * 192 + r], v);
  }
}

// ---------------------------------------------------------------------------
// Kernel 3: p1t1 = sum_x p3n^2 + p1n ; p3t1 = p3n * p1t1
// ---------------------------------------------------------------------------
__global__ __launch_bounds__(256) void finalize_kernel(
    const float* __restrict__ p1acc, const float* __restrict__ p3acc,
    float* __restrict__ out) {
  int e = blockIdx.x * 256 + threadIdx.x;          // atom*64 + c
  if (e >= NATOMS * FDIM) return;
  int a = e >> 6, c = e & 63;
  float x0 = p3acc[(size_t)a * 192 + c];
  float x1 = p3acc[(size_t)a * 192 + 64 + c];
  float x2 = p3acc[(size_t)a * 192 + 128 + c];
  float t = x0 * x0 + x1 * x1 + x2 * x2 + p1acc[e];
  out[e] = t;
  float* o3 = out + (size_t)NATOMS * FDIM;
  o3[(size_t)a * 192 + c]        = x0 * t;
  o3[(size_t)a * 192 + 64 + c]   = x1 * t;
  o3[(size_t)a * 192 + 128 + c]  = x2 * t;
}

extern "C" void kernel_launch(void* const* d_in, const int* in_sizes, int n_in,
                              void* d_out, int out_size, void* d_ws, size_t ws_size,
                              hipStream_t stream) {
  const float*     p1    = (const float*)d_in[0];
  const float*     p3    = (const float*)d_in[1];
  const float*     basis = (const float*)d_in[2];
  const float*     diff  = (const float*)d_in[3];
  const float*     w0    = (const float*)d_in[4];
  const float*     b0    = (const float*)d_in[5];
  const float*     w1    = (const float*)d_in[6];
  const float*     b1    = (const float*)d_in[7];
  const float*     piw   = (const float*)d_in[8];
  const float*     pib   = (const float*)d_in[9];
  const float*     iiw   = (const float*)d_in[10];
  const long long* ind2  = (const long long*)d_in[11];
  float* out = (float*)d_out;

  float* h     = (float*)d_ws;                       // 50000*64
  float* p1acc = h + (size_t)NATOMS * FDIM;          // 50000*64
  float* p3acc = p1acc + (size_t)NATOMS * FDIM;      // 50000*192

  // zero the accumulators every call (harness does not re-poison)
  hipMemsetAsync(p1acc, 0, (size_t)NATOMS * FDIM * 4 * sizeof(float), stream);

  atom_mlp_kernel<<<NATOMS / 16, 128, 0, stream>>>(p1, w0, b0, w1, b1, h);
  pair_kernel<<<NPAIRS / 16, 128, 0, stream>>>(h, p3, basis, diff, piw, pib, iiw,
                                               ind2, p1acc, p3acc);
  finalize_kernel<<<(NATOMS * FDIM + 255) / 256, 256, 0, stream>>>(p1acc, p3acc, out);
}